// ControllerLayer_56332791054964
// MI455X (gfx1250) — compile-verified
//
#include <hip/hip_runtime.h>
#include <hip/hip_bf16.h>

// Problem sizes (fixed by the reference)
#define BB 8
#define SS 1024
#define DD 1024
#define EE 1024
#define MM (BB * SS)      // 8192 tokens
#define ALPHA_F 0.01f

typedef __attribute__((ext_vector_type(2))) float v2f;
typedef __attribute__((ext_vector_type(8))) float v8f;

// ---------------------------------------------------------------------------
// Kernel 0: zero expert counts
// ---------------------------------------------------------------------------
__global__ void moe_init_counts(int* __restrict__ counts) {
    int i = blockIdx.x * blockDim.x + threadIdx.x;
    if (i < EE) counts[i] = 0;
}

// ---------------------------------------------------------------------------
// Kernel 1: router GEMM  logits[m,e] = sum_d x[m,d] * W[e,d]
// fp32 WMMA 16x16x4. Workgroup = 256 threads (8 waves), WG tile 128(M)x128(N),
// wave tile 32(M)x64(N) = 2x4 WMMA tiles.
// A fragment (16x4 f32): lane<16 -> rows M=lane, K={k,k+1}; lane>=16 -> K={k+2,k+3}
// B fragment (4x16 f32): lane<16 -> cols N=lane, K={k,k+1}; lane>=16 -> K={k+2,k+3}
// Both are contiguous float2 loads from row-major x / W.
// ---------------------------------------------------------------------------
__global__ __launch_bounds__(256)
void moe_router_gemm(const float* __restrict__ x,
                     const float* __restrict__ W,
                     float* __restrict__ logits) {
    const int tid    = threadIdx.x;
    const int wave   = tid >> 5;
    const int lane   = tid & 31;
    const int lane16 = lane & 15;
    const int hi     = (lane >> 4) & 1;

    const int wm = wave & 3;   // 0..3  -> M sub-tile
    const int wn = wave >> 2;  // 0..1  -> N sub-tile

    const int m0 = blockIdx.y * 128 + wm * 32;  // wave M base (2 tiles of 16)
    const int n0 = blockIdx.x * 128 + wn * 64;  // wave N base (4 tiles of 16)

    v8f acc[2][4];
#pragma unroll
    for (int mt = 0; mt < 2; ++mt)
#pragma unroll
        for (int nt = 0; nt < 4; ++nt)
            acc[mt][nt] = (v8f){0.f, 0.f, 0.f, 0.f, 0.f, 0.f, 0.f, 0.f};

    // Per-lane fragment base pointers (advance by k along the row)
    const float* ap[2];
    const float* bp[4];
#pragma unroll
    for (int mt = 0; mt < 2; ++mt)
        ap[mt] = x + (size_t)(m0 + mt * 16 + lane16) * DD + 2 * hi;
#pragma unroll
    for (int nt = 0; nt < 4; ++nt)
        bp[nt] = W + (size_t)(n0 + nt * 16 + lane16) * DD + 2 * hi;

#pragma unroll 2
    for (int k = 0; k < DD; k += 4) {
        v2f a[2], b[4];
#pragma unroll
        for (int mt = 0; mt < 2; ++mt)
            a[mt] = *(const v2f*)(ap[mt] + k);
#pragma unroll
        for (int nt = 0; nt < 4; ++nt)
            b[nt] = *(const v2f*)(bp[nt] + k);

#pragma unroll
        for (int mt = 0; mt < 2; ++mt)
#pragma unroll
            for (int nt = 0; nt < 4; ++nt)
                acc[mt][nt] = __builtin_amdgcn_wmma_f32_16x16x4_f32(
                    /*neg_a=*/false, a[mt],
                    /*neg_b=*/false, b[nt],
                    /*c_mod=*/(short)0, acc[mt][nt],
                    /*reuse_a=*/false, /*reuse_b=*/false);
    }

    // C/D layout: VGPR r: lanes 0-15 -> M=r, N=lane; lanes 16-31 -> M=8+r, N=lane-16
#pragma unroll
    for (int mt = 0; mt < 2; ++mt) {
#pragma unroll
        for (int nt = 0; nt < 4; ++nt) {
            const int row = m0 + mt * 16 + hi * 8;
            const int col = n0 + nt * 16 + lane16;
            float* out = logits + (size_t)row * EE + col;
#pragma unroll
            for (int r = 0; r < 8; ++r)
                out[(size_t)r * EE] = acc[mt][nt][r];
        }
    }
}

// ---------------------------------------------------------------------------
// Kernel 2: per-token softmax + top-2. One block (256 thr) per token row.
// Overwrites logits with probs in-place. Tie-break: lowest index (= jax top_k).
// ---------------------------------------------------------------------------
__global__ __launch_bounds__(256)
void moe_softmax_top2(float* __restrict__ logits_probs,
                      float* __restrict__ topk_p,
                      int* __restrict__ topk_i,
                      int* __restrict__ counts) {
    const int row  = blockIdx.x;          // 0..MM-1
    const int tid  = threadIdx.x;         // 0..255
    float* rowp = logits_probs + (size_t)row * EE;

    __shared__ float sv[256];
    __shared__ int   si[256];

    float l[4];
#pragma unroll
    for (int j = 0; j < 4; ++j) l[j] = rowp[tid + 256 * j];

    // ---- pass 1: argmax (gives row max for softmax and top-1 index) ----
    float bv = l[0];
    int   bi = tid;
#pragma unroll
    for (int j = 1; j < 4; ++j) {
        const int idx = tid + 256 * j;
        if (l[j] > bv || (l[j] == bv && idx < bi)) { bv = l[j]; bi = idx; }
    }
    sv[tid] = bv; si[tid] = bi;
    __syncthreads();
    for (int s = 128; s > 0; s >>= 1) {
        if (tid < s) {
            const float ov = sv[tid + s];
            const int   oi = si[tid + s];
            if (ov > sv[tid] || (ov == sv[tid] && oi < si[tid])) { sv[tid] = ov; si[tid] = oi; }
        }
        __syncthreads();
    }
    const float m1 = sv[0];
    const int   i1 = si[0];
    __syncthreads();

    // ---- exp + sum ----
    float e[4];
    float ls = 0.f;
#pragma unroll
    for (int j = 0; j < 4; ++j) { e[j] = __expf(l[j] - m1); ls += e[j]; }
    sv[tid] = ls;
    __syncthreads();
    for (int s = 128; s > 0; s >>= 1) {
        if (tid < s) sv[tid] += sv[tid + s];
        __syncthreads();
    }
    const float sum  = sv[0];
    __syncthreads();
    const float inv  = 1.0f / sum;

    // write probs (overwrite logits)
#pragma unroll
    for (int j = 0; j < 4; ++j) rowp[tid + 256 * j] = e[j] * inv;

    // ---- pass 2: argmax excluding i1 (on exp values; monotone in logits) ----
    float bv2 = -1.0f;
    int   bi2 = 0x7fffffff;
#pragma unroll
    for (int j = 0; j < 4; ++j) {
        const int idx = tid + 256 * j;
        const float v = (idx == i1) ? -1.0f : e[j];
        if (v > bv2 || (v == bv2 && idx < bi2)) { bv2 = v; bi2 = idx; }
    }
    sv[tid] = bv2; si[tid] = bi2;
    __syncthreads();
    for (int s = 128; s > 0; s >>= 1) {
        if (tid < s) {
            const float ov = sv[tid + s];
            const int   oi = si[tid + s];
            if (ov > sv[tid] || (ov == sv[tid] && oi < si[tid])) { sv[tid] = ov; si[tid] = oi; }
        }
        __syncthreads();
    }
    if (tid == 0) {
        const float e2 = sv[0];
        const int   i2 = si[0];
        topk_p[2 * row]     = inv;       // exp(m1-m1)/sum = 1/sum
        topk_p[2 * row + 1] = e2 * inv;
        topk_i[2 * row]     = i1;
        topk_i[2 * row + 1] = i2;
        atomicAdd(&counts[i1], 1);       // integer atomics: deterministic
        atomicAdd(&counts[i2], 1);
    }
}

// ---------------------------------------------------------------------------
// Kernel 3: deterministic column partial-sums of probs.
// 64 blocks, each reduces a 128-token chunk over all 1024 experts (float4 rows).
// ---------------------------------------------------------------------------
__global__ __launch_bounds__(256)
void moe_colsum(const float* __restrict__ probs, float* __restrict__ partial) {
    const int chunk = blockIdx.x;   // 0..63
    const int tid   = threadIdx.x;  // 0..255 -> experts tid*4..tid*4+3
    float a0 = 0.f, a1 = 0.f, a2 = 0.f, a3 = 0.f;
    const int t0 = chunk * 128;
    for (int t = t0; t < t0 + 128; ++t) {
        const float4 p = *(const float4*)(probs + (size_t)t * EE + tid * 4);
        a0 += p.x; a1 += p.y; a2 += p.z; a3 += p.w;
    }
    float4 o; o.x = a0; o.y = a1; o.z = a2; o.w = a3;
    *(float4*)(partial + (size_t)chunk * EE + tid * 4) = o;
}

// ---------------------------------------------------------------------------
// Kernel 4: combine  y[t,:] = p0 * x[b,i0,:] + p1 * x[b,i1,:]
// One block (256 thr) per token; float4 per thread.
// ---------------------------------------------------------------------------
__global__ __launch_bounds__(256)
void moe_combine(const float* __restrict__ x,
                 const float* __restrict__ topk_p,
                 const int* __restrict__ topk_i,
                 float* __restrict__ y) {
    const int t = blockIdx.x;       // token
    const int b = t >> 10;          // S = 1024
    const float p0 = topk_p[2 * t];
    const float p1 = topk_p[2 * t + 1];
    const int   i0 = topk_i[2 * t];
    const int   i1 = topk_i[2 * t + 1];
    const float4* r0 = (const float4*)(x + ((size_t)b * SS + i0) * DD);
    const float4* r1 = (const float4*)(x + ((size_t)b * SS + i1) * DD);
    const float4 v0 = r0[threadIdx.x];
    const float4 v1 = r1[threadIdx.x];
    float4 o;
    o.x = p0 * v0.x + p1 * v1.x;
    o.y = p0 * v0.y + p1 * v1.y;
    o.z = p0 * v0.z + p1 * v1.z;
    o.w = p0 * v0.w + p1 * v1.w;
    ((float4*)(y + (size_t)t * DD))[threadIdx.x] = o;
}

// ---------------------------------------------------------------------------
// Kernel 5: aux loss. Single block.
// aux = ALPHA*E * sum_e (router_probs[e] * counts[e]/16384), router_probs = colmean
// ---------------------------------------------------------------------------
__global__ __launch_bounds__(256)
void moe_aux(const float* __restrict__ partial,
             const int* __restrict__ counts,
             float* __restrict__ aux_out) {
    const int tid = threadIdx.x;
    float acc = 0.f;
#pragma unroll
    for (int j = 0; j < 4; ++j) {
        const int e = tid * 4 + j;
        float rs = 0.f;
        for (int c = 0; c < 64; ++c) rs += partial[(size_t)c * EE + e];
        acc += rs * (float)counts[e];
    }
    __shared__ float sv[256];
    sv[tid] = acc;
    __syncthreads();
    for (int s = 128; s > 0; s >>= 1) {
        if (tid < s) sv[tid] += sv[tid + s];
        __syncthreads();
    }
    if (tid == 0) {
        const float scale = (float)(0.01 * 1024.0 / (8192.0 * 16384.0));
        aux_out[0] = scale * sv[0];
    }
}

// ---------------------------------------------------------------------------
// Host launcher
// ---------------------------------------------------------------------------
extern "C" void kernel_launch(void* const* d_in, const int* in_sizes, int n_in,
                              void* d_out, int out_size, void* d_ws, size_t ws_size,
                              hipStream_t stream) {
    const float* x = (const float*)d_in[0];   // (B,S,D) fp32
    const float* W = (const float*)d_in[1];   // (E,D)   fp32
    float* y   = (float*)d_out;               // (B,S,D) flat
    float* aux = y + (size_t)MM * DD;         // scalar appended

    // Workspace layout
    float* logits  = (float*)d_ws;                       // MM*EE  (becomes probs)
    float* topk_p  = logits + (size_t)MM * EE;           // 2*MM
    int*   topk_i  = (int*)(topk_p + 2 * MM);            // 2*MM
    int*   counts  = topk_i + 2 * MM;                    // EE
    float* partial = (float*)(counts + EE);              // 64*EE

    (void)in_sizes; (void)n_in; (void)out_size; (void)ws_size;

    moe_init_counts<<<(EE + 255) / 256, 256, 0, stream>>>(counts);

    dim3 gemm_grid(EE / 128, MM / 128);   // (8, 64)
    moe_router_gemm<<<gemm_grid, 256, 0, stream>>>(x, W, logits);

    moe_softmax_top2<<<MM, 256, 0, stream>>>(logits, topk_p, topk_i, counts);

    moe_colsum<<<64, 256, 0, stream>>>(logits, partial);

    moe_combine<<<MM, 256, 0, stream>>>(x, topk_p, topk_i, y);

    moe_aux<<<1, 256, 0, stream>>>(partial, counts, aux);
}